// Heston_52862457479520
// MI455X (gfx1250) — compile-verified
//
#include <hip/hip_runtime.h>
#include <hip/hip_bf16.h>
#include <math.h>

// ---------------------------------------------------------------------------
// Heston Euler-Maruyama path simulation for MI455X (gfx1250).
// Memory-bound streaming kernel (~1.03 GB @ 23.3 TB/s ~= 44us floor).
// CDNA5 path used: GLOBAL_LOAD_ASYNC_TO_LDS_B128 (ASYNCcnt, double-buffered
// input staging) + LDS-staged coalesced B128 output stores.
// ---------------------------------------------------------------------------

#define N_PATHS  100000
#define N_STEPS  512
#define TPB      128          // 4 waves of 32
#define CHUNK    16           // time steps per staged tile
#define NCHUNK   (N_STEPS / CHUNK)

// LDS row geometry (bytes / floats)
#define ZROW_F   36           // 16 steps * 2 floats = 32 payload + 4 pad (144B, 16B aligned)
#define OROW_F   52           // 16 steps * 3 floats = 48 payload + 4 pad (208B, 16B aligned)

#if defined(__has_builtin)
# if __has_builtin(__builtin_amdgcn_global_load_async_to_lds_b128)
#  define USE_ASYNC_BUILTIN 1
# endif
# if __has_builtin(__builtin_amdgcn_s_wait_asynccnt)
#  define USE_WAITASYNC_BUILTIN 1
# endif
#endif

typedef int v4i __attribute__((ext_vector_type(4)));
typedef __attribute__((address_space(1))) v4i as1_v4i;   // global
typedef __attribute__((address_space(3))) v4i as3_v4i;   // LDS

__device__ __forceinline__ void async_load16(const void* g, void* l) {
#ifdef USE_ASYNC_BUILTIN
  void* gnc = const_cast<void*>(g);
  __builtin_amdgcn_global_load_async_to_lds_b128((as1_v4i*)gnc, (as3_v4i*)l, 0, 0);
#else
  // vdst = LDS byte offset (low 32 bits of generic LDS address), vaddr = 64-bit global
  asm volatile("global_load_async_to_lds_b128 %0, %1, off"
               :: "v"((unsigned)(unsigned long long)l),
                  "v"((unsigned long long)g)
               : "memory");
#endif
}

template <int N>
__device__ __forceinline__ void wait_async() {
#ifdef USE_WAITASYNC_BUILTIN
  __builtin_amdgcn_s_wait_asynccnt(N);
#else
  if (N == 0) asm volatile("s_wait_asynccnt 0" ::: "memory");
  else        asm volatile("s_wait_asynccnt 8" ::: "memory");
#endif
  asm volatile("" ::: "memory");   // compiler barrier: no LDS read hoisting
}

__global__ __launch_bounds__(TPB)
void heston_paths_kernel(const float* __restrict__ z, float* __restrict__ out) {
  __shared__ float zb[2][TPB * ZROW_F];   // 2 x 18,432 B
  __shared__ float ob[TPB * OROW_F];      // 26,624 B   (total 63,488 B)

  const int tid  = threadIdx.x;
  const int wave = tid >> 5;
  const int lane = tid & 31;
  const int blockbase = blockIdx.x * TPB;
  const int wavebase  = blockbase + wave * 32;
  if (wavebase >= N_PATHS) return;        // uniform per wave (N_PATHS % 32 == 0)

  const int p = blockbase + tid;          // this thread's path

  // --- constants -----------------------------------------------------------
  const float DT    = 0.004f;
  const float KAPPA = 1.0f;
  const float THETA = 0.04f;
  const float SIGMA = 0.2f;
  const float RHO   = -0.7f;
  const float S0    = 100.0f;
  const float V0    = 0.04f;
  const float SQDT  = 0.063245553203367586f;            // sqrt(DT)
  const float CSQDT = 0.7141428428542850f * SQDT;       // sqrt(1-rho^2)*sqrt(DT)
  const float KDT   = KAPPA * DT;

  const char* zbytes = (const char*)z;

  // --- async tile prefetch: wave w stages its own 32 paths x CHUNK steps ---
  // global: path stride 4096B, chunk k at +k*128, 8 contiguous 16B segments.
  // LDS:    row stride 144B (16B aligned), 8 segments of 16B.
  auto issue_loads = [&](int k, int b) {
    const int q   = lane & 7;        // segment within path
    const int pw0 = lane >> 3;       // 4 paths per 32-lane issue
#pragma unroll
    for (int r = 0; r < 8; ++r) {
      const int row = wave * 32 + pw0 + r * 4;
      const void* g = zbytes + (size_t)(blockbase + row) * 4096
                             + (size_t)k * 128 + (size_t)q * 16;
      void* l = (char*)&zb[b][0] + row * 144 + q * 16;
      async_load16(g, l);
    }
  };

  issue_loads(0, 0);

  // --- t = 0 column (tiny, direct store: 12B per path, once) ---------------
  {
    const float tau0 = (float)N_STEPS * DT;
    const float L0 = (V0 - THETA) / KAPPA * (1.0f - __expf(-KAPPA * tau0)) + THETA * tau0;
    const float w0 = V0 * DT + L0;
    float* o0 = out + (size_t)p * (3 * (N_STEPS + 1));
    o0[0] = S0; o0[1] = V0; o0[2] = w0;
  }

  float s = S0, v = V0, acc = V0;   // acc = running cumsum of variances (incl. v0)

  for (int k = 0; k < NCHUNK; ++k) {
    const int b = k & 1;
    if (k + 1 < NCHUNK) { issue_loads(k + 1, b ^ 1); wait_async<8>(); }
    else                { wait_async<0>(); }

    const float* zrow = &zb[b][tid * ZROW_F];
    float*       orow = &ob[tid * OROW_F];

#pragma unroll
    for (int tl = 0; tl < CHUNK; ++tl) {
      const float z0 = zrow[2 * tl + 0];
      const float z1 = zrow[2 * tl + 1];
      const float vp  = fmaxf(v, 0.0f);
      const float sv  = sqrtf(vp);
      const float dW1 = SQDT * z0;
      const float dW2 = fmaf(CSQDT, z1, RHO * dW1);
      s = fmaf(s * sv, dW1, s);                               // s += s*sv*dW1
      v = fmaxf(fmaf(SIGMA * sv, dW2, fmaf(KDT, THETA - vp, v)), 0.0f);
      acc += v;
      const int   tg  = k * CHUNK + tl + 1;                   // output time index
      const float tau = (float)(N_STEPS - tg) * DT;
      const float L   = fmaf(v - THETA, 1.0f - __expf(-KAPPA * tau), THETA * tau);
      const float wsw = fmaf(acc, DT, L);
      orow[3 * tl + 0] = s;
      orow[3 * tl + 1] = v;
      orow[3 * tl + 2] = wsw;
    }

    // --- drain out tile, coalesced b128: 32 paths * 12 segments of 16B -----
    // DS ops are in-order per wave => no barrier needed (wave owns its rows).
#pragma unroll
    for (int r = 0; r < 12; ++r) {
      const int idx = r * 32 + lane;
      const int pw  = idx / 12;
      const int q   = idx - pw * 12;
      const int row = wave * 32 + pw;
      const float4 val = *(const float4*)&ob[row * OROW_F + q * 4];
      float* gdst = out + (size_t)(blockbase + row) * (3 * (N_STEPS + 1))
                        + k * (CHUNK * 3) + 3 + q * 4;
      *(float4*)gdst = val;                                   // global_store_b128
    }
  }
}

extern "C" void kernel_launch(void* const* d_in, const int* in_sizes, int n_in,
                              void* d_out, int out_size, void* d_ws, size_t ws_size,
                              hipStream_t stream) {
  (void)in_sizes; (void)n_in; (void)out_size; (void)d_ws; (void)ws_size;
  const float* z = (const float*)d_in[0];
  float* out = (float*)d_out;
  const int grid = (N_PATHS + TPB - 1) / TPB;   // 782 blocks
  heston_paths_kernel<<<grid, TPB, 0, stream>>>(z, out);
}